// LeafCompletionNetwork_60876866453859
// MI455X (gfx1250) — compile-verified
//
#include <hip/hip_runtime.h>

// ---------------- problem constants (match the JAX reference) ----------------
#define GAP_R    0.05f
#define GAP_R2   (GAP_R * GAP_R)
#define STRIDE_C 10
#define MIN_LEAF 10
#define NB       8
#define NPTS     8192
#define NFEAT    256
#define MAXC     (NPTS / STRIDE_C + 1)   // 820 candidate slots (reference maxc)
#define MAXC_PAD 832                     // padded to 52 tiles of 16
#define CTILES   (MAXC_PAD / 16)         // 52 candidate tiles
#define PTILES   (NPTS / 16)             // 512 point tiles
#define RCAP     2048                    // ring list capacity (expected ~15)
#define WCAP     512                     // write list capacity (expected ~2)

typedef __attribute__((ext_vector_type(2))) float v2f;
typedef __attribute__((ext_vector_type(8))) float v8f;

// -----------------------------------------------------------------------------
// Kernel 1: per-batch candidate selection.
// rank = cumsum(mask)-1 ; candidate iff mask && rank%10==0 ; slot = rank/10.
// Stores centers as float4 (x,y,z,|c|^2); invalid slots pushed far away.
// -----------------------------------------------------------------------------
__global__ void cand_kernel(const float* __restrict__ pts,          // [B,N,3]
                            const unsigned char* __restrict__ mask, // [B,N]
                            int* __restrict__ cand_idx,             // [B,MAXC_PAD]
                            float4* __restrict__ centers,           // [B,MAXC_PAD]
                            int* __restrict__ leaf_cnt)             // [B]
{
    const int b = blockIdx.x;
    const float*         P  = pts  + (size_t)b * NPTS * 3;
    const unsigned char* M  = mask + (size_t)b * NPTS;
    int*    CI = cand_idx + b * MAXC_PAD;
    float4* CT = centers  + b * MAXC_PAD;

    __shared__ int part[256];
    __shared__ int scan[256];
    const int t = threadIdx.x;

    for (int i = t; i < MAXC_PAD; i += 256) {
        CI[i] = -1;
        CT[i] = make_float4(1e9f, 1e9f, 1e9f, 3e18f);  // never within GAP_R
    }

    const int CH = NPTS / 256;     // 32 points per thread, contiguous chunk
    const int base = t * CH;
    int cnt = 0;
    for (int i = 0; i < CH; ++i) cnt += (M[base + i] ? 1 : 0);
    part[t] = cnt;
    __syncthreads();

    if (t == 0) {
        int run = 0;
        for (int i = 0; i < 256; ++i) { scan[i] = run; run += part[i]; }
        leaf_cnt[b] = run;
    }
    __syncthreads();

    int rank = scan[t];
    for (int i = 0; i < CH; ++i) {
        const int gi = base + i;
        if (M[gi]) {
            if (rank % STRIDE_C == 0) {
                const int slot = rank / STRIDE_C;   // <= 819 < MAXC_PAD
                CI[slot] = gi;
                const float x = P[gi * 3 + 0];
                const float y = P[gi * 3 + 1];
                const float z = P[gi * 3 + 2];
                CT[slot] = make_float4(x, y, z, x * x + y * y + z * z);
            }
            ++rank;
        }
    }
}

// -----------------------------------------------------------------------------
// Kernel 2: gap detection via V_WMMA_F32_16X16X4_F32.
// d^2(c,p) = |c|^2 + |p|^2 - 2*(c.p) ; c.p computed as a 16x16x4 tile matmul
// (K = x,y,z,0). Block = 16 candidates x all points; 8 waves, each owning the
// point tiles {wave + 8*j}, unrolled x4 so 16 independent loads are in flight
// and 4 WMMAs issue per iteration (A operand is loop-invariant).
// A layout (ISA 7.12.2): lanes 0-15 hold K0/K1 in vgpr0/1, lanes 16-31 K2/K3.
// C layout: vgpr v -> row v + 8*(lane>=16), col = lane&15.
// -----------------------------------------------------------------------------
__global__ void gap_kernel(const float* __restrict__ pts,
                           const unsigned char* __restrict__ mask,
                           const int* __restrict__ cand_idx,
                           const float4* __restrict__ centers,
                           const int* __restrict__ leaf_cnt,
                           int* __restrict__ is_gap)
{
    const int tile = blockIdx.x;   // candidate tile (0..51)
    const int b    = blockIdx.y;
    const float*         P = pts  + (size_t)b * NPTS * 3;
    const unsigned char* M = mask + (size_t)b * NPTS;
    const float4* CT = centers + b * MAXC_PAD + tile * 16;

    __shared__ int cnt16[16];
    const int t = threadIdx.x;
    if (t < 16) cnt16[t] = 0;
    __syncthreads();

    const int lane = t & 31;
    const int wave = t >> 5;                 // 0..7
    const int half = (lane < 16) ? 0 : 1;
    const int col  = lane & 15;
    const int rowbase = half * 8;

    // A matrix: 16 candidate rows, K = (x,y,z,0)
    const float4 cme = CT[col];
    v2f a;
    a.x = half ? cme.z : cme.x;              // K2 : K0
    a.y = half ? 0.0f  : cme.y;              // K3 : K1

    // |c|^2 for the 8 rows this lane's half covers in the C accumulator
    float cn[8];
#pragma unroll
    for (int v = 0; v < 8; ++v) cn[v] = CT[rowbase + v].w;

    int cacc[8] = {0, 0, 0, 0, 0, 0, 0, 0};

    const int JT = PTILES / 8;               // 64 tiles per wave
    for (int j = 0; j < JT; j += 4) {
        float px[4], py[4], pz[4];
        int   pm[4];
#pragma unroll
        for (int u = 0; u < 4; ++u) {
            const int pt = wave + (j + u) * 8;
            const int gi = pt * 16 + col;    // both halves load the same 16 pts
            px[u] = P[gi * 3 + 0];
            py[u] = P[gi * 3 + 1];
            pz[u] = P[gi * 3 + 2];
            pm[u] = M[gi] ? 1 : 0;
        }
        // prefetch the next unroll group of this wave
        {
            const int ptn = wave + (j + 4) * 8;
            __builtin_prefetch(P + (size_t)(ptn * 16 + col) * 3, 0, 1);
        }
#pragma unroll
        for (int u = 0; u < 4; ++u) {
            const float pn = px[u] * px[u] + py[u] * py[u] + pz[u] * pz[u];
            v2f bm;
            bm.x = half ? pz[u] : px[u];     // K2 : K0
            bm.y = half ? 0.0f  : py[u];     // K3 : K1

            v8f acc = {};
            acc = __builtin_amdgcn_wmma_f32_16x16x4_f32(
                      false, a, false, bm, (short)0, acc, false, false);

            if (pm[u]) {
#pragma unroll
                for (int v = 0; v < 8; ++v) {
                    const float d2 = cn[v] + pn - 2.0f * acc[v];
                    cacc[v] += (d2 < GAP_R2) ? 1 : 0;
                }
            }
        }
    }

#pragma unroll
    for (int v = 0; v < 8; ++v)
        if (cacc[v]) atomicAdd(&cnt16[rowbase + v], cacc[v]);
    __syncthreads();

    if (t < 16) {
        const int c  = tile * 16 + t;
        const int ci = cand_idx[b * MAXC_PAD + c];
        const int lc = leaf_cnt[b];
        is_gap[b * MAXC_PAD + c] =
            (ci >= 0) && (lc >= MIN_LEAF) && (cnt16[t] < 3) ? 1 : 0;
    }
}

// -----------------------------------------------------------------------------
// Kernel 3: features -> d_out (fill mutates d_out in place afterwards).
// -----------------------------------------------------------------------------
__global__ void copy_kernel(const float4* __restrict__ src,
                            float4* __restrict__ dst, int n4)
{
    int i = blockIdx.x * blockDim.x + threadIdx.x;
    const int stride = gridDim.x * blockDim.x;
    for (; i < n4; i += stride) dst[i] = src[i];
}

// -----------------------------------------------------------------------------
// Kernel 4: sequential gap fill (one block per batch; scan order = reference).
// Deterministic compaction: per-thread counts -> serial exclusive scan ->
// in-order placement, so the ring feature sum is in ascending point order
// every run. Thread t owns feature dim t (NFEAT == blockDim) -> coalesced.
// -----------------------------------------------------------------------------
__global__ void fill_kernel(const float* __restrict__ pts,
                            const unsigned char* __restrict__ mask,
                            const float4* __restrict__ centers,
                            const int* __restrict__ is_gap,
                            float* __restrict__ F)   // [B,N,D] in d_out
{
    const int b = blockIdx.x;
    const float*         P = pts  + (size_t)b * NPTS * 3;
    const unsigned char* M = mask + (size_t)b * NPTS;
    const float4* CT = centers + b * MAXC_PAD;
    const int*    G  = is_gap  + b * MAXC_PAD;
    float* Fb = F + (size_t)b * NPTS * NFEAT;

    __shared__ int rpart[256], wpart[256], rbase[256], wbase[256];
    __shared__ int ringCnt, wrCnt;
    __shared__ int ringList[RCAP];
    __shared__ int wrList[WCAP];

    const int t  = threadIdx.x;          // 256
    const int CH = NPTS / 256;           // 32, contiguous chunk per thread

    for (int c = 0; c < MAXC_PAD; ++c) {
        const float4 ctr = CT[c];
        const int    gap = G[c];

        // pass 1: count ring / write hits in this thread's chunk
        int rc = 0, wc = 0;
        for (int i = 0; i < CH; ++i) {
            const int gi = t * CH + i;
            const float dx = P[gi * 3 + 0] - ctr.x;
            const float dy = P[gi * 3 + 1] - ctr.y;
            const float dz = P[gi * 3 + 2] - ctr.z;
            const float d2 = dx * dx + dy * dy + dz * dz;
            if (d2 <= GAP_R2)                      ++wc;            // dist <= R
            else if (d2 < 4.0f * GAP_R2 && M[gi])  ++rc;            // R < dist < 2R
        }
        rpart[t] = rc;
        wpart[t] = wc;
        __syncthreads();

        if (t == 0) {
            int rr = 0, wr = 0;
            for (int i = 0; i < 256; ++i) {
                rbase[i] = rr; rr += rpart[i];
                wbase[i] = wr; wr += wpart[i];
            }
            ringCnt = rr; wrCnt = wr;
        }
        __syncthreads();

        // pass 2: in-order placement (list sorted by point index -> deterministic)
        int rp = rbase[t], wp = wbase[t];
        for (int i = 0; i < CH; ++i) {
            const int gi = t * CH + i;
            const float dx = P[gi * 3 + 0] - ctr.x;
            const float dy = P[gi * 3 + 1] - ctr.y;
            const float dz = P[gi * 3 + 2] - ctr.z;
            const float d2 = dx * dx + dy * dy + dz * dz;
            if (d2 <= GAP_R2) {
                if (wp < WCAP) wrList[wp] = gi;
                ++wp;
            } else if (d2 < 4.0f * GAP_R2 && M[gi]) {
                if (rp < RCAP) ringList[rp] = gi;
                ++rp;
            }
        }
        __syncthreads();

        const int rcnt = ringCnt;
        if (gap && rcnt > 3) {
            const int nr = rcnt < RCAP ? rcnt : RCAP;
            float s = 0.0f;
            for (int i = 0; i < nr; ++i)
                s += Fb[(size_t)ringList[i] * NFEAT + t];
            const float avg = s / (float)rcnt;
            const int nw = wrCnt < WCAP ? wrCnt : WCAP;
            for (int i = 0; i < nw; ++i)
                Fb[(size_t)wrList[i] * NFEAT + t] = avg;
            __threadfence_block();    // later candidates must see these writes
        }
        __syncthreads();
    }
}

// -----------------------------------------------------------------------------
extern "C" void kernel_launch(void* const* d_in, const int* in_sizes, int n_in,
                              void* d_out, int out_size, void* d_ws, size_t ws_size,
                              hipStream_t stream)
{
    (void)in_sizes; (void)n_in; (void)out_size; (void)ws_size;

    const float*         pts   = (const float*)d_in[0];          // [B,N,3]
    const float*         feats = (const float*)d_in[1];          // [B,N,D]
    const unsigned char* mask  = (const unsigned char*)d_in[2];  // [B,N] bool (1B)
    float* out = (float*)d_out;

    char*  ws  = (char*)d_ws;
    size_t off = 0;
    int* cand_idx = (int*)(ws + off);  off += (size_t)NB * MAXC_PAD * sizeof(int);
    off = (off + 15) & ~(size_t)15;
    float4* centers = (float4*)(ws + off); off += (size_t)NB * MAXC_PAD * sizeof(float4);
    int* is_gap   = (int*)(ws + off);  off += (size_t)NB * MAXC_PAD * sizeof(int);
    int* leaf_cnt = (int*)(ws + off);  off += (size_t)NB * sizeof(int);

    cand_kernel<<<NB, 256, 0, stream>>>(pts, mask, cand_idx, centers, leaf_cnt);
    gap_kernel<<<dim3(CTILES, NB), 256, 0, stream>>>(pts, mask, cand_idx, centers,
                                                     leaf_cnt, is_gap);
    const int n4 = NB * NPTS * NFEAT / 4;
    copy_kernel<<<1024, 256, 0, stream>>>((const float4*)feats, (float4*)out, n4);
    fill_kernel<<<NB, 256, 0, stream>>>(pts, mask, centers, is_gap, out);
}